// MyNet_31748398252694
// MI455X (gfx1250) — compile-verified
//
#include <hip/hip_runtime.h>
#include <hip/hip_bf16.h>

typedef __attribute__((ext_vector_type(2))) float v2f;
typedef __attribute__((ext_vector_type(8))) float v8f;

// ---------------------------------------------------------------------------
// Implicit-GEMM 3x3 conv (stride 1, pad 1) via V_WMMA_F32_16X16X4_F32.
//   O[COUT, B*H*W] = Wt[COUT, CIN*9] x im2col(in)[CIN*9, B*H*W]
// Block: one (b, y) output row, WTILE x-pixels, all COUT.
// LDS:
//   s_wt : zero-padded weights [COUT][KPAD] (pairs along K are adjacent)
//   s_b2 : im2col chunk, K-pair interleaved: s_b2[(kk>>1)*WTILE*2 + x*2 + (kk&1)]
// Inner loop: per 4-K step, (MPW + XPW) ds_load_b64 with *immediate* offsets
// + MPW*XPW wmma. H/W are compile-time so all global addressing is
// 32-bit shifts + immediate store offsets (no 64-bit mul chains in the loop).
// ---------------------------------------------------------------------------
template <int CIN, int COUT, int WTILE, int XPW, int MPW, int KCHUNK, int H, int W>
__global__ __launch_bounds__((WTILE / (16 * XPW)) * (COUT / (16 * MPW)) * 32)
void conv3x3_wmma_f32(const float* __restrict__ in, const float* __restrict__ wgt,
                      float* __restrict__ out) {
  constexpr int K    = CIN * 9;
  constexpr int KPAD = (K + 3) & ~3;
  constexpr int NW_X = WTILE / (16 * XPW);
  constexpr int NW_C = COUT / (16 * MPW);
  constexpr int NTHR = NW_X * NW_C * 32;
  constexpr int HW   = H * W;
  static_assert(KPAD % KCHUNK == 0 && KCHUNK % 4 == 0, "bad KCHUNK");

  __shared__ float s_wt[COUT * KPAD];
  __shared__ float s_b2[KCHUNK * WTILE];

  const int tid  = threadIdx.x;
  const int lane = tid & 31;
  const int wave = tid >> 5;

  const int x0 = blockIdx.x * WTILE;
  const int y  = blockIdx.y;
  const int b  = blockIdx.z;

  // ---- stage weights once (zero-pad K -> KPAD) ----
#pragma clang loop unroll(disable)
  for (int co = 0; co < COUT; ++co) {
#pragma clang loop unroll(disable)
    for (int k = tid; k < KPAD; k += NTHR)
      s_wt[co * KPAD + k] = (k < K) ? wgt[co * K + k] : 0.0f;
  }

  const int n     = lane & 15;
  const int khalf = (lane >> 4) << 1;  // 0 or 2
  const int xw    = wave % NW_X;
  const int cw    = wave / NW_X;

  const float* abase = s_wt + (cw * MPW * 16 + n) * KPAD + khalf;
  const float* bbase = s_b2 + ((khalf >> 1) * WTILE + (xw * XPW) * 16 + n) * 2;

  v8f acc[MPW][XPW];
#pragma unroll
  for (int mi = 0; mi < MPW; ++mi)
#pragma unroll
    for (int xi = 0; xi < XPW; ++xi)
      acc[mi][xi] = (v8f){0.f, 0.f, 0.f, 0.f, 0.f, 0.f, 0.f, 0.f};

  const float* inb = in + (unsigned)(b * CIN) * HW;  // batch base (32-bit offsets below)

  for (int kc = 0; kc < KPAD; kc += KCHUNK) {
    __syncthreads();  // previous chunk fully consumed

    // ---- stage im2col chunk (pair-interleaved), division-free inner loop ----
#pragma clang loop unroll(disable)
    for (int kk = 0; kk < KCHUNK; ++kk) {
      const int k   = kc + kk;
      const int row = k / 3;           // = ci*3 + r   (uniform per iteration)
      const int s   = k - row * 3;
      const int ci  = row / 3;
      const int r   = row - ci * 3;
      const int gy  = y + r - 1;
      const bool rowok = (k < K) & (gy >= 0) & (gy < H);
      const int gyc = gy < 0 ? 0 : (gy >= H ? H - 1 : gy);
      const float* src = inb + (unsigned)((ci < CIN ? ci : 0) * H + gyc) * W;
      float* dst = s_b2 + (kk >> 1) * (WTILE * 2) + (kk & 1);
#pragma clang loop unroll(disable)
      for (int x = tid; x < WTILE; x += NTHR) {
        const int gx = x0 + x + s - 1;
        float v = (rowok && gx >= 0 && gx < W) ? src[gx] : 0.0f;
        dst[2 * x] = v;
      }
    }
    __syncthreads();

    // ---- compute: fully unrolled, immediate-offset LDS loads ----
#pragma unroll
    for (int k0 = 0; k0 < KCHUNK; k0 += 4) {
      v2f a[MPW];
#pragma unroll
      for (int mi = 0; mi < MPW; ++mi)
        a[mi] = *(const v2f*)(abase + mi * 16 * KPAD + kc + k0);
      v2f bf[XPW];
#pragma unroll
      for (int xi = 0; xi < XPW; ++xi)
        bf[xi] = *(const v2f*)(bbase + k0 * WTILE + xi * 32);
#pragma unroll
      for (int mi = 0; mi < MPW; ++mi)
#pragma unroll
        for (int xi = 0; xi < XPW; ++xi)
          acc[mi][xi] = __builtin_amdgcn_wmma_f32_16x16x4_f32(
              false, a[mi], false, bf[xi], (short)0, acc[mi][xi], false, false);
    }
  }

  // ---- epilogue: C/D layout lane l, vgpr v -> M = v + (l>=16)*8, N = l&15 ----
  // 32-bit flat offsets; per-channel stride HW*4 bytes becomes an immediate.
  const int mbase = (lane >> 4) * 8;
  float* outb = out + (unsigned)(b * COUT) * HW + (unsigned)y * W;
#pragma unroll
  for (int mi = 0; mi < MPW; ++mi) {
#pragma unroll
    for (int xi = 0; xi < XPW; ++xi) {
      const int gx = x0 + (xw * XPW + xi) * 16 + n;
      float* p = outb + (unsigned)(((cw * MPW + mi) * 16 + mbase) * HW) + gx;
#pragma unroll
      for (int v = 0; v < 8; ++v)
        p[v * HW] = acc[mi][xi][v];
    }
  }
}

// ---------------------------------------------------------------------------
// BatchNorm (training): per-channel mean + rsqrt(var+eps). One block / channel.
// B=16, C=64, HW=65536 hardcoded (shifts, no divides).
// ---------------------------------------------------------------------------
__global__ __launch_bounds__(1024)
void bn_stats_kernel(const float* __restrict__ h, float* __restrict__ stats) {
  constexpr int B = 16, C = 64, HW = 1 << 16;
  const int c   = blockIdx.x;
  const int tid = threadIdx.x;
  const int nt  = 1024;

  float sum = 0.f, sumsq = 0.f;
#pragma clang loop unroll(disable)
  for (int b = 0; b < B; ++b) {
    const float4* p = (const float4*)(h + ((unsigned)(b * C + c) << 16));
#pragma clang loop unroll(disable)
    for (int i = tid; i < (HW >> 2); i += nt) {
      float4 v = p[i];
      sum   += v.x + v.y + v.z + v.w;
      sumsq += v.x * v.x + v.y * v.y + v.z * v.z + v.w * v.w;
    }
  }

  __shared__ float ssum[1024];
  __shared__ float ssq[1024];
  ssum[tid] = sum;
  ssq[tid]  = sumsq;
  __syncthreads();
#pragma clang loop unroll(disable)
  for (int s = nt >> 1; s > 0; s >>= 1) {
    if (tid < s) {
      ssum[tid] += ssum[tid + s];
      ssq[tid]  += ssq[tid + s];
    }
    __syncthreads();
  }
  if (tid == 0) {
    const float N    = (float)B * (float)HW;
    const float mean = ssum[0] / N;
    const float var  = ssq[0] / N - mean * mean;
    stats[c]     = mean;
    stats[C + c] = rsqrtf(var + 1e-5f);
  }
}

// ---------------------------------------------------------------------------
// In-place normalize: out = (h - mean) * rstd * gamma + beta  (float4 lanes).
// Grid covers exactly B*C*HW/4 = 16.78M float4 -> no bounds guard.
// Channel = (elem >> 16) & 63 since HW = 65536, C = 64.
// ---------------------------------------------------------------------------
__global__ __launch_bounds__(256)
void bn_apply_kernel(float* __restrict__ h, const float* __restrict__ stats,
                     const float* __restrict__ gamma, const float* __restrict__ beta) {
  const unsigned i = blockIdx.x * 256u + threadIdx.x;   // < 2^24
  const int c = (int)((i >> 14) & 63);                  // (i*4 >> 16) & 63
  const float rstd = stats[64 + c];
  const float g    = gamma[c] * rstd;
  const float bb   = beta[c] - stats[c] * g;
  float4* p = (float4*)h;
  float4 v = p[i];
  v.x = v.x * g + bb;
  v.y = v.y * g + bb;
  v.z = v.z * g + bb;
  v.w = v.w * g + bb;
  p[i] = v;
}

extern "C" void kernel_launch(void* const* d_in, const int* in_sizes, int n_in,
                              void* d_out, int out_size, void* d_ws, size_t ws_size,
                              hipStream_t stream) {
  const float* x     = (const float*)d_in[0];
  const float* w1    = (const float*)d_in[1];
  const float* w2    = (const float*)d_in[2];
  const float* w3    = (const float*)d_in[3];
  const float* gamma = (const float*)d_in[4];
  const float* beta  = (const float*)d_in[5];
  float* out = (float*)d_out;

  constexpr int B = 16, H = 256, W = 256;
  const size_t plane16 = (size_t)B * 16 * H * W;

  float* h1    = (float*)d_ws;
  float* h2    = h1 + plane16;
  float* stats = h2 + plane16;  // 128 floats (mean[64], rstd[64])

  // conv1: Cin=3  -> 16 : WTILE=128, 2 waves, K=27 (KPAD=28, 1 chunk)
  conv3x3_wmma_f32<3, 16, 128, 4, 1, 28, H, W>
      <<<dim3(W / 128, H, B), 64, 0, stream>>>(x, w1, h1);
  // conv2: 16 -> 16 : WTILE=128, 2 waves, K=144 in 2 chunks of 72
  conv3x3_wmma_f32<16, 16, 128, 4, 1, 72, H, W>
      <<<dim3(W / 128, H, B), 64, 0, stream>>>(h1, w2, h2);
  // conv3: 16 -> 64 : WTILE=64, 2 waves (2 co-tiles each), K=144 in 2 chunks
  conv3x3_wmma_f32<16, 64, 64, 4, 2, 72, H, W>
      <<<dim3(W / 64, H, B), 64, 0, stream>>>(h2, w3, out);

  bn_stats_kernel<<<64, 1024, 0, stream>>>(out, stats);

  // 16*64*65536/4 float4 elems / 256 threads = 65536 blocks
  bn_apply_kernel<<<65536, 256, 0, stream>>>(out, stats, gamma, beta);
}